// pano_att_gcn_v6_82179904242096
// MI455X (gfx1250) — compile-verified
//
#include <hip/hip_runtime.h>
#include <hip/hip_bf16.h>
#include <math.h>

#define QDIM 512
#define CTX  1576
#define NVW  36
#define BA   1024
#define HDIM 2200   // CTX + QDIM + 112
#define QKLD 1600   // 1576 padded to multiple of 32

typedef __attribute__((ext_vector_type(2))) float v2f;
typedef __attribute__((ext_vector_type(8))) float v8f;

#if defined(__has_builtin)
#if __has_builtin(__builtin_amdgcn_global_load_async_to_lds_b128) && \
    __has_builtin(__builtin_amdgcn_s_wait_asynccnt)
#define USE_ASYNC_LDS 1
#endif
#endif

// ---------------------------------------------------------------------------
// fp32 WMMA GEMM: C = op(A[M,K] * B), optional tanh epilogue.
// One wave per 32x32 tile = 2x2 WMMA 16x16 accumulators; each A/B fragment
// feeds two v_wmma_f32_16x16x4_f32 ops (halves load traffic per FLOP).
// A fragment (16x4): lanes 0-15 -> K={k,k+1}, lanes 16-31 -> K={k+2,k+3}.
// B fragment (4x16): mirrored. C/D: vgpr r -> row r (+8 for lanes>=16).
// ---------------------------------------------------------------------------
template <bool TRANSB, bool TANH>
__global__ __launch_bounds__(256) void gemm_f32_kernel(
    const float* __restrict__ A, int lda,
    const float* __restrict__ B, int ldb,
    float* __restrict__ C, int ldc,
    int M, int N, int K)
{
    const int wave   = (int)((blockIdx.x * blockDim.x + threadIdx.x) >> 5);
    const int ntiles = N >> 5;
    const int tm = wave / ntiles;
    const int tn = wave % ntiles;
    if (tm >= (M >> 5)) return;               // whole-wave uniform exit
    const int m0 = tm << 5, n0 = tn << 5;

    const int lane = threadIdx.x & 31;
    const int half = lane >> 4;   // 0: K pair {0,1}, 1: K pair {2,3}
    const int l16  = lane & 15;
    const int am0  = m0 + l16;
    const int am1  = m0 + 16 + l16;
    const int bn0  = n0 + l16;
    const int bn1  = n0 + 16 + l16;

    v8f acc00 = {}, acc01 = {}, acc10 = {}, acc11 = {};
    for (int k = 0; k < K; k += 4) {
        const int ka = k + half * 2;
        v2f a0, a1, b0, b1;
        a0.x = A[(size_t)am0 * lda + ka];
        a0.y = A[(size_t)am0 * lda + ka + 1];
        a1.x = A[(size_t)am1 * lda + ka];
        a1.y = A[(size_t)am1 * lda + ka + 1];
        if (TRANSB) {
            b0.x = B[(size_t)bn0 * ldb + ka];
            b0.y = B[(size_t)bn0 * ldb + ka + 1];
            b1.x = B[(size_t)bn1 * ldb + ka];
            b1.y = B[(size_t)bn1 * ldb + ka + 1];
        } else {
            b0.x = B[(size_t)(ka)     * ldb + bn0];
            b0.y = B[(size_t)(ka + 1) * ldb + bn0];
            b1.x = B[(size_t)(ka)     * ldb + bn1];
            b1.y = B[(size_t)(ka + 1) * ldb + bn1];
        }
        acc00 = __builtin_amdgcn_wmma_f32_16x16x4_f32(false, a0, false, b0, (short)0, acc00, false, false);
        acc01 = __builtin_amdgcn_wmma_f32_16x16x4_f32(false, a0, false, b1, (short)0, acc01, false, false);
        acc10 = __builtin_amdgcn_wmma_f32_16x16x4_f32(false, a1, false, b0, (short)0, acc10, false, false);
        acc11 = __builtin_amdgcn_wmma_f32_16x16x4_f32(false, a1, false, b1, (short)0, acc11, false, false);
    }

    const int rb0 = m0 + half * 8;
    const int rb1 = m0 + 16 + half * 8;
    const int c0  = n0 + l16;
    const int c1  = n0 + 16 + l16;
#pragma unroll
    for (int r = 0; r < 8; ++r) {
        float v00 = acc00[r], v01 = acc01[r], v10 = acc10[r], v11 = acc11[r];
        if (TANH) { v00 = tanhf(v00); v01 = tanhf(v01); v10 = tanhf(v10); v11 = tanhf(v11); }
        C[(size_t)(rb0 + r) * ldc + c0] = v00;
        C[(size_t)(rb0 + r) * ldc + c1] = v01;
        C[(size_t)(rb1 + r) * ldc + c0] = v10;
        C[(size_t)(rb1 + r) * ldc + c1] = v11;
    }
}

// ---------------------------------------------------------------------------
// Pano affinity table (device-side, deterministic, float64 math like numpy).
// ---------------------------------------------------------------------------
__global__ void pano_kernel(float* __restrict__ out)
{
    if (threadIdx.x != 0 || blockIdx.x != 0) return;
    double a[NVW][NVW];
    for (int i = 0; i < NVW; ++i)
        for (int j = 0; j < NVW; ++j) a[i][j] = (i == j) ? 1.0 : 0.0;
    for (int v = 0; v < 12; ++v) {
        a[v][v + 12] = 1;
        if (v == 0)  { a[0][11] = 1; a[0][23] = 1; }
        else         { a[v][v - 1] = 1; a[v][v + 11] = 1; }
        if (v == 11) { a[11][0] = 1; a[11][12] = 1; }
        else         { a[v][v + 1] = 1; a[v][v + 13] = 1; }
    }
    for (int v = 12; v < 24; ++v) {
        a[v][v + 12] = 1; a[v][v - 12] = 1;
        if (v == 12) { a[12][23] = 1; a[12][35] = 1; a[12][11] = 1; }
        else         { a[v][v - 1] = 1; a[v][v + 11] = 1; a[v][v - 13] = 1; }
        if (v == 23) { a[23][12] = 1; a[23][24] = 1; a[23][0] = 1; }
        else         { a[v][v + 1] = 1; a[v][v + 13] = 1; a[v][v - 11] = 1; }
    }
    for (int v = 24; v < 36; ++v) {
        a[v][v - 12] = 1;
        if (v == 24) { a[24][35] = 1; a[24][23] = 1; }
        else         { a[v][v - 1] = 1; a[v][v - 13] = 1; }
        if (v == 35) { a[35][24] = 1; a[35][12] = 1; }
        else         { a[v][v + 1] = 1; a[v][v - 11] = 1; }
    }
    double h[NVW][34];
    for (int r = 0; r < NVW; ++r)
        for (int c = 0; c < 34; ++c)
            h[r][c] = 0.25 * a[r][c] + 0.5 * a[r][c + 1] + 0.25 * a[r][c + 2];
    float o[NVW][NVW];
    for (int i = 0; i < NVW; ++i)
        for (int j = 0; j < NVW; ++j) o[i][j] = (float)a[i][j];
    for (int i = 0; i < 34; ++i)
        for (int j = 0; j < 34; ++j)
            o[i + 1][j + 1] = (float)(0.25 * h[i][j] + 0.5 * h[i + 1][j] + 0.25 * h[i + 2][j]);
    for (int i = 0; i < NVW; ++i) o[i][i] = 1.0f;
    for (int i = 0; i < NVW; ++i)
        for (int j = 0; j < NVW; ++j) {
            float x = o[i][j];
            out[i * NVW + j] = (x == 0.0f) ? 0.01f : x;
        }
}

// ---------------------------------------------------------------------------
// Fused attention: one block per batch-row b.
//  - stage V[b] (36x1576 f32 = 227KB) + qk[b] in LDS; V is read from HBM once.
//    Uses GLOBAL_LOAD_ASYNC_TO_LDS_B128 (ASYNCcnt) when the builtin exists.
//  - dq[b] = dot(dproj[b], Qf[b]) via wave 0
//  - energy[n] = dot(V[b,n,:], qk[b,:]) + dq   -> softmax -> attn (output)
//  - attn_gcn = attn * pano[tid[b]]
//  - wc[c] = sum_n attn_gcn[n] * V[b,n,c]; write h = [wc | Qf | kv]
// ---------------------------------------------------------------------------
__global__ __launch_bounds__(256) void fused_attn_kernel(
    const float* __restrict__ V,
    const float* __restrict__ qk, int ldqk,
    const float* __restrict__ Qf,
    const float* __restrict__ dproj,
    const float* __restrict__ kv,
    const int*   __restrict__ tid_arr,
    const float* __restrict__ pano,
    float* __restrict__ h,
    float* __restrict__ attn_out)
{
    const int b    = blockIdx.x;
    const int tid  = threadIdx.x;
    const int lane = tid & 31;
    const int w    = tid >> 5;

    extern __shared__ float lds[];
    float* sV  = lds;              // NVW*CTX
    float* sqk = lds + NVW * CTX;  // CTX
    __shared__ float s_en[NVW];
    __shared__ float s_attn[NVW];
    __shared__ float s_dq;

    // Stage V[b] into LDS with 128-bit transfers (CTX % 4 == 0).
    const float4* Vg  = (const float4*)(V + (size_t)b * NVW * CTX);
    float4*       sV4 = (float4*)sV;
#ifdef USE_ASYNC_LDS
    typedef int b128_t __attribute__((vector_size(16)));
    typedef __attribute__((address_space(1))) b128_t* gptr_b128;
    typedef __attribute__((address_space(3))) b128_t* lptr_b128;
    for (int i = tid; i < (NVW * CTX) / 4; i += 256)
        __builtin_amdgcn_global_load_async_to_lds_b128(
            (gptr_b128)(Vg + i), (lptr_b128)(sV4 + i), 0, 0);
#else
    for (int i = tid; i < (NVW * CTX) / 4; i += 256) sV4[i] = Vg[i];
#endif
    for (int i = tid; i < CTX; i += 256) sqk[i] = qk[(size_t)b * ldqk + i];

    if (w == 0) {  // dq = dot(dproj[b], Qf[b])
        float p = 0.f;
        for (int q = lane; q < QDIM; q += 32)
            p += dproj[(size_t)b * QDIM + q] * Qf[(size_t)b * QDIM + q];
        for (int o = 16; o > 0; o >>= 1) p += __shfl_xor(p, o, 32);
        if (lane == 0) s_dq = p;
    }
#ifdef USE_ASYNC_LDS
    __builtin_amdgcn_s_wait_asynccnt(0);
#endif
    __syncthreads();

    const float scale = 0.04419417382415922f;  // 512^-0.5
    for (int n = w; n < NVW; n += 8) {
        float p = 0.f;
        for (int c = lane; c < CTX; c += 32) p += sV[n * CTX + c] * sqk[c];
        for (int o = 16; o > 0; o >>= 1) p += __shfl_xor(p, o, 32);
        if (lane == 0) s_en[n] = (p + s_dq) * scale;
    }
    __syncthreads();

    if (tid == 0) {  // tiny 36-way softmax, serial
        const int t = tid_arr[b];
        float mx = s_en[0];
        for (int n = 1; n < NVW; ++n) mx = fmaxf(mx, s_en[n]);
        float e[NVW], sum = 0.f;
        for (int n = 0; n < NVW; ++n) { e[n] = expf(s_en[n] - mx); sum += e[n]; }
        const float inv = 1.f / sum;
        for (int n = 0; n < NVW; ++n) {
            const float att = e[n] * inv;
            attn_out[(size_t)b * NVW + n] = att;
            s_attn[n] = att * pano[t * NVW + n];
        }
    }
    __syncthreads();

    float* hrow = h + (size_t)b * HDIM;
    for (int c = tid; c < CTX; c += 256) {
        float acc = 0.f;
#pragma unroll
        for (int n = 0; n < NVW; ++n) acc = fmaf(s_attn[n], sV[n * CTX + c], acc);
        hrow[c] = acc;
    }
    for (int q = tid; q < QDIM; q += 256) hrow[CTX + q] = Qf[(size_t)b * QDIM + q];
    for (int i = tid; i < 112;  i += 256) hrow[CTX + QDIM + i] = kv[(size_t)b * 112 + i];
}

// ---------------------------------------------------------------------------
extern "C" void kernel_launch(void* const* d_in, const int* in_sizes, int n_in,
                              void* d_out, int out_size, void* d_ws, size_t ws_size,
                              hipStream_t stream)
{
    (void)in_sizes; (void)n_in; (void)out_size; (void)ws_size;
    const float* Q   = (const float*)d_in[0];   // (1024, 512)
    const float* V   = (const float*)d_in[1];   // (1024, 36, 1576)
    const float* df  = (const float*)d_in[2];   // (1024, 900)
    const float* kv  = (const float*)d_in[3];   // (1024, 112)
    const int*   tid = (const int*)  d_in[4];   // (1024,)
    const float* Wk  = (const float*)d_in[5];   // (512, 2476)
    const float* Wq  = (const float*)d_in[6];   // (512, 512)
    const float* Wo  = (const float*)d_in[7];   // (512, 2200)
    float* out = (float*)d_out;                 // [h_tilde 1024*512 | attn 1024*36]
    float* ws  = (float*)d_ws;

    float* Qf    = ws;                          // 1024*512
    float* qk    = Qf    + (size_t)BA * QDIM;   // 1024*1600 (padded)
    float* dproj = qk    + (size_t)BA * QKLD;   // 1024*512
    float* pano  = dproj + (size_t)BA * QDIM;   // 36*36 (+pad)
    float* h     = pano  + 1312;                // 1024*2200

    const size_t lds_bytes = (size_t)(NVW * CTX + CTX) * sizeof(float);
    (void)hipFuncSetAttribute((const void*)fused_attn_kernel,
                              hipFuncAttributeMaxDynamicSharedMemorySize,
                              (int)lds_bytes);

    pano_kernel<<<1, 32, 0, stream>>>(pano);

    // Qf = Q @ Wq^T : M=1024 N=512 K=512 -> 512 waves
    gemm_f32_kernel<true, false><<<64, 256, 0, stream>>>(
        Q, QDIM, Wq, QDIM, Qf, QDIM, BA, QDIM, QDIM);

    // qk = Qf @ Wk[:, :1576] (padded N=1600) : 32*50 = 1600 waves
    gemm_f32_kernel<false, false><<<200, 256, 0, stream>>>(
        Qf, QDIM, Wk, 2476, qk, QKLD, BA, QKLD, QDIM);

    // dproj = df @ Wk[:, 1576:]^T : M=1024 N=512 K=900 -> 512 waves
    gemm_f32_kernel<true, false><<<64, 256, 0, stream>>>(
        df, 900, Wk + CTX, 2476, dproj, QDIM, BA, QDIM, 900);

    // fused attention: V staged in LDS (read once), writes attn + h rows
    fused_attn_kernel<<<BA, 256, lds_bytes, stream>>>(
        V, qk, QKLD, Qf, dproj, kv, tid, pano, h, out + (size_t)BA * QDIM);

    // h_tilde = tanh(h @ Wo^T) : M=1024 N=512 K=2200 -> 512 waves
    gemm_f32_kernel<true, true><<<64, 256, 0, stream>>>(
        h, HDIM, Wo, HDIM, out, QDIM, BA, QDIM, HDIM);
}